// Point_contextual_hyper_prior_encoder_6047313953085
// MI455X (gfx1250) — compile-verified
//
#include <hip/hip_runtime.h>

// ---------------------------------------------------------------------------
// Sparse conv net (gather -> GEMM -> scatter) on MI455X / gfx1250.
// fp32 end-to-end via V_WMMA_F32_16X16X4_F32 (exact fp32 accumulate, matrix
// pipe engaged). Working set (~240MB) is mostly resident in the 192MB L2;
// compute (~7.6 GFLOP) is far below any ceiling, so the kernel is organized
// around the gather / scatter-atomic path, not the MACs.
//   grid = (chunk blocks, col-tile, offset)  -> no div/mod wave decode
//   B fragments register-resident per wave across 8 pair-blocks
//   full blocks: unguarded b64 gathers + 2x b128 out-index loads
// ---------------------------------------------------------------------------

typedef __attribute__((ext_vector_type(2))) float v2f;
typedef __attribute__((ext_vector_type(8))) float v8f;
typedef __attribute__((ext_vector_type(4))) int v4i;

__global__ void zero_f32(float* __restrict__ p, long long n) {
    long long i = (long long)blockIdx.x * blockDim.x + threadIdx.x;
    long long stride = (long long)gridDim.x * blockDim.x;
    for (; i < n; i += stride) p[i] = 0.0f;
}

// y rows are 64 wide; apply bias + ReLU in place to first n elements.
__global__ void bias_relu64(float* __restrict__ y, const float* __restrict__ b,
                            long long n) {
    long long i = (long long)blockIdx.x * blockDim.x + threadIdx.x;
    long long stride = (long long)gridDim.x * blockDim.x;
    for (; i < n; i += stride) {
        float v = y[i] + b[(int)(i & 63)];
        y[i] = v > 0.0f ? v : 0.0f;
    }
}

// out = acc + b3 over n2*8 elements (dummy row of acc dropped).
__global__ void bias_out8(const float* __restrict__ acc,
                          const float* __restrict__ b,
                          float* __restrict__ out, long long n) {
    long long i = (long long)blockIdx.x * blockDim.x + threadIdx.x;
    long long stride = (long long)gridDim.x * blockDim.x;
    for (; i < n; i += stride) out[i] = acc[i] + b[(int)(i & 7)];
}

// ---------------------------------------------------------------------------
// Gather-WMMA-scatter sparse conv.
//   blockIdx.z = kernel offset k, blockIdx.y = 16-col tile ct,
//   (blockIdx.x, wave-in-block) = chunk of BPC pair-blocks of 16 pairs.
//   A (16x4 f32): lane m = lane&15 owns pair-row m; lanes 0-15 carry K0/K1,
//   lanes 16-31 carry K2/K3 (ISA 7.12.2). D (16x16 f32, 8 VGPRs): lane&15 =
//   N column, vgpr index (+8 for hi lanes) = M row -> per-element
//   global_atomic_add_f32 scatter (unique rows within an offset; atomics
//   arbitrate only cross-offset collisions).
// ---------------------------------------------------------------------------
template <int CIN, int COUT, int KOFF>
__global__ void sparse_conv_wmma(const float* __restrict__ x,
                                 const float* __restrict__ W,
                                 const int* __restrict__ in_idx,
                                 const int* __restrict__ out_idx,
                                 float* __restrict__ yacc, // (n_out+1) x COUT
                                 int L, int blocksPerChunk) {
    const int lane = threadIdx.x & 31;
    const int k = blockIdx.z;
    const int ct = blockIdx.y;
    const int chunk = blockIdx.x * (blockDim.x >> 5) + (threadIdx.x >> 5);

    const bool hiHalf = lane >= 16;
    const int m = lane & 15;
    const int col = ct * 16 + m;
    const bool colValid = col < COUT;

    // --- B fragments for (k, ct): register-resident across the chunk -------
    const float* Wk = W + (long long)k * CIN * COUT;
    v2f bfrag[CIN / 4];
#pragma unroll
    for (int kk = 0; kk < CIN / 4; ++kk) {
        const int krow = kk * 4 + (hiHalf ? 2 : 0);
        float b0 = 0.0f, b1 = 0.0f;
        if (colValid) {
            b0 = Wk[(krow + 0) * COUT + col];
            b1 = Wk[(krow + 1) * COUT + col];
        }
        bfrag[kk].x = b0;
        bfrag[kk].y = b1;
    }

    const int* __restrict__ inK = in_idx + (long long)k * L;
    const int* __restrict__ outK = out_idx + (long long)k * L;
    const int mbase = hiHalf ? 8 : 0;

    for (int bblk = 0; bblk < blocksPerChunk; ++bblk) {
        const int p0 = (chunk * blocksPerChunk + bblk) * 16;
        if (p0 >= L) break; // uniform per wave

        // stream-prefetch upcoming map entries (sequential within a chunk)
        __builtin_prefetch(inK + p0 + 16 * blocksPerChunk, 0, 0);
        __builtin_prefetch(outK + p0 + 16 * blocksPerChunk, 0, 0);

        v8f c = {};
        if (p0 + 16 <= L) {
            // ---------------- fast path: full 16-row block ----------------
            const int inRow = inK[p0 + m];
            const float* __restrict__ xr = x + (long long)inRow * CIN;
#pragma unroll
            for (int kk = 0; kk < CIN / 4; ++kk) {
                const int kc = kk * 4 + (hiHalf ? 2 : 0);
                const v2f a = *(const v2f*)(xr + kc); // global_load_b64
                c = __builtin_amdgcn_wmma_f32_16x16x4_f32(
                    false, a, false, bfrag[kk], (short)0, c, false, false);
            }
            if (colValid) {
                const v4i o0 = *(const v4i*)(outK + p0 + mbase);     // b128
                const v4i o1 = *(const v4i*)(outK + p0 + mbase + 4); // b128
                const int orow[8] = {o0.x, o0.y, o0.z, o0.w,
                                     o1.x, o1.y, o1.z, o1.w};
#pragma unroll
                for (int r = 0; r < 8; ++r)
                    unsafeAtomicAdd(&yacc[(long long)orow[r] * COUT + col],
                                    c[r]);
            }
        } else {
            // ---------------- tail path: guarded partial block ------------
            const int p = p0 + m;
            const bool pvalid = p < L;
            const int inRow = pvalid ? inK[p] : 0;
            const float* __restrict__ xr = x + (long long)inRow * CIN;
#pragma unroll
            for (int kk = 0; kk < CIN / 4; ++kk) {
                const int kc = kk * 4 + (hiHalf ? 2 : 0);
                v2f a;
                a.x = pvalid ? xr[kc + 0] : 0.0f;
                a.y = pvalid ? xr[kc + 1] : 0.0f;
                c = __builtin_amdgcn_wmma_f32_16x16x4_f32(
                    false, a, false, bfrag[kk], (short)0, c, false, false);
            }
            if (colValid) {
#pragma unroll
                for (int r = 0; r < 8; ++r) {
                    const int pm = p0 + mbase + r;
                    if (pm < L) {
                        const int orow = outK[pm];
                        unsafeAtomicAdd(
                            &yacc[(long long)orow * COUT + col], c[r]);
                    }
                }
            }
        }
    }
}

template <int CIN, int COUT, int KOFF>
static void launch_conv(const float* x, const float* W, const int* ii,
                        const int* oi, float* y, int L, hipStream_t s) {
    constexpr int NCT = (COUT + 15) / 16;
    const int BPC = 8; // 8 pair-blocks (128 pairs) per wave amortizes B load
    const int blocks16 = (L + 15) / 16;
    const int nChunks = (blocks16 + BPC - 1) / BPC;
    dim3 grid((nChunks + 7) / 8, NCT, KOFF); // 8 waves / 256-thread block
    sparse_conv_wmma<CIN, COUT, KOFF>
        <<<grid, 256, 0, s>>>(x, W, ii, oi, y, L, BPC);
}

extern "C" void kernel_launch(void* const* d_in, const int* in_sizes, int n_in,
                              void* d_out, int out_size, void* d_ws,
                              size_t ws_size, hipStream_t stream) {
    const float* feats = (const float*)d_in[0];
    const float* W1 = (const float*)d_in[1];
    const float* b1 = (const float*)d_in[2];
    const float* W2 = (const float*)d_in[3];
    const float* b2 = (const float*)d_in[4];
    const float* W3 = (const float*)d_in[5];
    const float* b3 = (const float*)d_in[6];
    const int* m1i = (const int*)d_in[7];
    const int* m1o = (const int*)d_in[8];
    const int* m2i = (const int*)d_in[9];
    const int* m2o = (const int*)d_in[10];
    const int* m3i = (const int*)d_in[11];
    const int* m3o = (const int*)d_in[12];
    // d_in[13] = n2 (device scalar) -- recovered host-side instead:

    const int N1 = in_sizes[0] / 8;   // input points
    const int L1 = in_sizes[7] / 27;  // pad length conv1 (== N1)
    const int L2 = in_sizes[9] / 8;   // pad length conv2
    const int L3 = in_sizes[11] / 27; // pad length conv3 == n2 (center offset
                                      // matches every coarse output row)
    const int n2 = L3;

    float* h1 = (float*)d_ws;                    // (N1+1) x 64
    const long long sz1 = (long long)(N1 + 1) * 64;
    float* h2 = h1 + sz1;                        // (n2+1) x 64
    const long long sz2 = (long long)(n2 + 1) * 64;
    float* acc3 = h2 + sz2;                      // (n2+1) x 8
    const long long sz3 = (long long)(n2 + 1) * 8;

    // zero all three accumulators in one sweep (they are contiguous)
    zero_f32<<<4096, 256, 0, stream>>>(h1, sz1 + sz2 + sz3);

    // conv1: 8 -> 64, 27 offsets, ReLU
    launch_conv<8, 64, 27>(feats, W1, m1i, m1o, h1, L1, stream);
    bias_relu64<<<4096, 256, 0, stream>>>(h1, b1, (long long)N1 * 64);

    // conv2: 64 -> 64 (stride-2 downsample), 8 offsets, ReLU
    launch_conv<64, 64, 8>(h1, W2, m2i, m2o, h2, L2, stream);
    bias_relu64<<<4096, 256, 0, stream>>>(h2, b2, (long long)n2 * 64);

    // conv3: 64 -> 8, 27 offsets on coarse lattice
    launch_conv<64, 8, 27>(h2, W3, m3i, m3o, acc3, L3, stream);

    bias_out8<<<2048, 256, 0, stream>>>(acc3, b3, (float*)d_out,
                                        (long long)n2 * 8);
}